// GCN_6090263626106
// MI455X (gfx1250) — compile-verified
//
#include <hip/hip_runtime.h>
#include <hip/hip_bf16.h>

typedef __attribute__((ext_vector_type(2))) float v2f;
typedef __attribute__((ext_vector_type(8))) float v8f;

#define NN 100000
#define NE 1600000
#define IN_DIM 256
#define HID_DIM 128
#define OUT_DIM 64

// ---------------- degree / symmetric norm ----------------
__global__ void k_deg_init(float* __restrict__ deg, int n) {
    int i = blockIdx.x * blockDim.x + threadIdx.x;
    if (i < n) deg[i] = 1.0f;  // self-loop contributes 1
}

__global__ void k_deg_edges(const int* __restrict__ dst, float* __restrict__ deg, int e) {
    int i = blockIdx.x * blockDim.x + threadIdx.x;
    if (i < e) atomicAdd(&deg[dst[i]], 1.0f);
}

__global__ void k_deg_to_dinv(float* __restrict__ deg, int n) {
    int i = blockIdx.x * blockDim.x + threadIdx.x;
    if (i < n) deg[i] = rsqrtf(fmaxf(deg[i], 1.0f));
}

// ---------------- fp32 WMMA GEMM:  H[M x K] * W[K x NOUT] -> H[M x NOUT] ----------------
// Compile-time K/NOUT (powers of two) -> no remainder loop, shift-only indexing.
// One block per 16-row panel of X; blockDim = 32*(NOUT/16); each wave owns one
// 16x16 output tile. 16xK X-panel is contiguous -> coalesced float4 stage into
// padded LDS (stride K+4, conflict-free A-fragment reads).
// 4 independent accumulators break the WMMA D->C serial dependency chain.
template <int K, int NOUT>
__launch_bounds__(32 * (NOUT / 16))
__global__ void k_gemm_wmma(const float* __restrict__ X, const float* __restrict__ W,
                            float* __restrict__ H) {
    constexpr int KP = K + 4;
    __shared__ float sX[16 * KP];

    const int    tm      = blockIdx.x;
    const size_t rowbase = (size_t)tm * 16 * (size_t)K;

    // cooperative coalesced stage of 16 full rows of X
    {
        const float4* g4 = (const float4*)(X + rowbase);
        constexpr int n4 = 4 * K;                 // (16*K)/4 float4s
        for (int i = threadIdx.x; i < n4; i += blockDim.x) {
            float4 v = g4[i];
            int fl = i << 2;                      // flat index in unpadded panel
            int r  = fl / K;                      // constexpr K -> shift
            int c  = fl & (K - 1);
            float* d = &sX[r * KP + c];
            d[0] = v.x; d[1] = v.y; d[2] = v.z; d[3] = v.w;
        }
    }
    __syncthreads();

    const int wave = threadIdx.x >> 5;
    const int lane = threadIdx.x & 31;
    const int half = lane >> 4;      // K-half selector per ISA A/B fragment layout
    const int l    = lane & 15;
    const int col0 = wave * 16;

    v8f acc0 = {}, acc1 = {}, acc2 = {}, acc3 = {};
    for (int k = 0; k < K; k += 16) {
        // 4 independent WMMA chains per iteration
        #pragma unroll
        for (int u = 0; u < 4; ++u) {
            const int kk = k + 4 * u + 2 * half;
            // A: lanes 0-15 hold K={kk,kk+1}, lanes 16-31 hold K={kk+2,kk+3}
            v2f a = *(const v2f*)&sX[l * KP + kk];
            // B: same K striping, column = col0 + l
            v2f b;
            b.x = W[kk * NOUT + col0 + l];
            b.y = W[(kk + 1) * NOUT + col0 + l];
            v8f& acc = (u == 0) ? acc0 : (u == 1) ? acc1 : (u == 2) ? acc2 : acc3;
            acc = __builtin_amdgcn_wmma_f32_16x16x4_f32(
                /*neg_a=*/false, a, /*neg_b=*/false, b,
                /*c_mod=*/(short)0, acc, /*reuse_a=*/false, /*reuse_b=*/false);
        }
    }
    v8f acc = (acc0 + acc1) + (acc2 + acc3);

    // D layout: VGPR r -> row (r + 8*half), col = l
    const size_t row0 = (size_t)tm * 16;
    #pragma unroll
    for (int r = 0; r < 8; ++r) {
        H[(row0 + r + 8 * half) * NOUT + col0 + l] = acc[r];
    }
}

// ---------------- edge gather/scale/scatter-add ----------------
// One float4 of features per thread; F/4 threads per edge (power of two).
__global__ void k_edge_agg(const int* __restrict__ src, const int* __restrict__ dst,
                           const float* __restrict__ dinv, const float* __restrict__ h,
                           float* __restrict__ agg, int e, int fshift, int F) {
    long long tid  = (long long)blockIdx.x * blockDim.x + threadIdx.x;
    long long eidx = tid >> fshift;
    int       f4   = (int)(tid & ((1 << fshift) - 1));
    if (eidx >= e) return;
    int s = src[eidx];
    int d = dst[eidx];
    float nrm = dinv[s] * dinv[d];
    float4 m = ((const float4*)(h + (size_t)s * F))[f4];
    float* o = agg + (size_t)d * F + ((size_t)f4 << 2);
    atomicAdd(o + 0, nrm * m.x);
    atomicAdd(o + 1, nrm * m.y);
    atomicAdd(o + 2, nrm * m.z);
    atomicAdd(o + 3, nrm * m.w);
}

// ---------------- layer-1 finalize: self-loop + bias + ReLU (in place) ----------------
__global__ void k_final_relu(float* __restrict__ agg, const float* __restrict__ hpre,
                             const float* __restrict__ dinv, const float* __restrict__ b,
                             long long total, int fshift) {
    long long i = (long long)blockIdx.x * blockDim.x + threadIdx.x;
    if (i >= total) return;
    int node = (int)(i >> fshift);
    int f    = (int)(i & ((1 << fshift) - 1));
    float self = dinv[node] * dinv[node];
    float v = agg[i] + self * hpre[i] + b[f];
    agg[i] = fmaxf(v, 0.0f);
}

// ---------------- layer-2 finalize + log_softmax (one wave per node, 64 classes) ---------
__global__ void k_final_logsoftmax(const float* __restrict__ agg, const float* __restrict__ hpre,
                                   const float* __restrict__ dinv, const float* __restrict__ b,
                                   float* __restrict__ out, int n) {
    int gtid = blockIdx.x * blockDim.x + threadIdx.x;
    int wid  = gtid >> 5;
    int lane = gtid & 31;
    if (wid >= n) return;
    float self = dinv[wid] * dinv[wid];
    size_t base = (size_t)wid * OUT_DIM;
    float v0 = agg[base + lane]      + self * hpre[base + lane]      + b[lane];
    float v1 = agg[base + lane + 32] + self * hpre[base + lane + 32] + b[lane + 32];
    float m = fmaxf(v0, v1);
    #pragma unroll
    for (int off = 16; off > 0; off >>= 1) m = fmaxf(m, __shfl_xor(m, off));
    float s = expf(v0 - m) + expf(v1 - m);
    #pragma unroll
    for (int off = 16; off > 0; off >>= 1) s += __shfl_xor(s, off);
    float lse = m + logf(s);
    out[base + lane]      = v0 - lse;
    out[base + lane + 32] = v1 - lse;
}

extern "C" void kernel_launch(void* const* d_in, const int* in_sizes, int n_in,
                              void* d_out, int out_size, void* d_ws, size_t ws_size,
                              hipStream_t stream) {
    const float* x  = (const float*)d_in[0];
    const int*   ei = (const int*)  d_in[1];
    const float* W1 = (const float*)d_in[2];
    const float* b1 = (const float*)d_in[3];
    const float* W2 = (const float*)d_in[4];
    const float* b2 = (const float*)d_in[5];
    float* out = (float*)d_out;

    const int N = NN, E = NE;
    const int* src = ei;       // edge_index[0]
    const int* dst = ei + E;   // edge_index[1]

    // workspace layout (floats): dinv | hpre1 | agg1 | hpre2 | agg2
    float* dinv  = (float*)d_ws;
    float* hpre1 = dinv  + N;
    float* agg1  = hpre1 + (size_t)N * HID_DIM;
    float* hpre2 = agg1  + (size_t)N * HID_DIM;
    float* agg2  = hpre2 + (size_t)N * OUT_DIM;

    // ---- symmetric normalization ----
    k_deg_init   <<<(N + 255) / 256, 256, 0, stream>>>(dinv, N);
    k_deg_edges  <<<(E + 255) / 256, 256, 0, stream>>>(dst, dinv, E);
    k_deg_to_dinv<<<(N + 255) / 256, 256, 0, stream>>>(dinv, N);

    // ---- layer 1: h = relu(Ahat (X W1) + b1) ----
    k_gemm_wmma<IN_DIM, HID_DIM><<<N / 16, 32 * (HID_DIM / 16), 0, stream>>>(x, W1, hpre1);
    hipMemsetAsync(agg1, 0, (size_t)N * HID_DIM * sizeof(float), stream);
    {
        long long work = (long long)E * (HID_DIM / 4);
        k_edge_agg<<<(unsigned)((work + 255) / 256), 256, 0, stream>>>(
            src, dst, dinv, hpre1, agg1, E, 5, HID_DIM);
        long long tot = (long long)N * HID_DIM;
        k_final_relu<<<(unsigned)((tot + 255) / 256), 256, 0, stream>>>(
            agg1, hpre1, dinv, b1, tot, 7);
    }

    // ---- layer 2: o = Ahat (h W2) + b2 ; log_softmax ----
    k_gemm_wmma<HID_DIM, OUT_DIM><<<N / 16, 32 * (OUT_DIM / 16), 0, stream>>>(agg1, W2, hpre2);
    hipMemsetAsync(agg2, 0, (size_t)N * OUT_DIM * sizeof(float), stream);
    {
        long long work = (long long)E * (OUT_DIM / 4);
        k_edge_agg<<<(unsigned)((work + 255) / 256), 256, 0, stream>>>(
            src, dst, dinv, hpre2, agg2, E, 4, OUT_DIM);
        k_final_logsoftmax<<<((N * 32) + 255) / 256, 256, 0, stream>>>(
            agg2, hpre2, dinv, b2, out, N);
    }
}